// FreqBiESN_31241592111410
// MI455X (gfx1250) — compile-verified
//
#include <hip/hip_runtime.h>
#include <cmath>

typedef float v2f __attribute__((ext_vector_type(2)));
typedef float v8f __attribute__((ext_vector_type(8)));

constexpr int BB = 128;    // batch
constexpr int LL = 2048;   // sequence length
constexpr int DD = 8;      // input dim
constexpr int HH = 32;     // hidden
constexpr int FEAT = 2176; // per-batch output features: 1024 + 1024 + 128

static __device__ __forceinline__ float rdlane(float v, int l) {
  return __builtin_bit_cast(float, __builtin_amdgcn_readlane(__builtin_bit_cast(int, v), l));
}

// ---------------------------------------------------------------------------
// Kernel 1: x_tr[b,l,j] = sum_d x[b,l,d] * W_in[d,j]
// ---------------------------------------------------------------------------
__global__ __launch_bounds__(256) void xtr_kernel(const float* __restrict__ x,
                                                  const float* __restrict__ W_in,
                                                  float* __restrict__ x_tr) {
  int gid = blockIdx.x * 256 + threadIdx.x;   // over BB*LL*HH
  int row = gid >> 5;
  int j   = gid & 31;
  const float* xr = x + (size_t)row * DD;
  float s = 0.f;
#pragma unroll
  for (int d = 0; d < DD; ++d) s += xr[d] * W_in[d * HH + j];
  x_tr[(size_t)row * HH + j] = s;
}

// ---------------------------------------------------------------------------
// Kernel 2: sequential ESN scan. One wave32 per (b, k, dir); lane j owns
// state element j and the j-th columns of W_s / W_c in VGPRs. Cross-lane
// broadcast via v_readlane, cyclic tap via a 256-deep LDS ring.
// ---------------------------------------------------------------------------
__global__ __launch_bounds__(32) void scan_kernel(const float* __restrict__ x_tr,
                                                  const float* __restrict__ W_fs,
                                                  const float* __restrict__ W_fc,
                                                  const float* __restrict__ W_bs,
                                                  const float* __restrict__ W_bc,
                                                  float* __restrict__ h_all) {
  int id  = blockIdx.x;          // 0..511
  int b   = id & (BB - 1);
  int k   = (id >> 7) & 1;
  int dir = id >> 8;
  int p   = (k == 0) ? 24 : 168;
  int j   = threadIdx.x;

  const float* Ws = dir ? W_bs : W_fs;
  const float* Wc = (dir ? W_bc : W_fc) + k * HH * HH;
  float wsc[HH], wcc[HH];
#pragma unroll
  for (int i = 0; i < HH; ++i) { wsc[i] = Ws[i * HH + j]; wcc[i] = Wc[i * HH + j]; }

  __shared__ float ring[256][HH];
  float hprev = 0.f;
  const size_t xbase = (size_t)b * LL * HH;
  const size_t hbase = (size_t)(b * 2 + k) * LL * 64 + dir * HH + j;

  for (int t = 0; t < LL; ++t) {
    int tt = dir ? (LL - 1 - t) : t;
    float xt = x_tr[xbase + (size_t)tt * HH + j];
    float hc = (t >= p) ? ring[(t - p) & 255][j] : 0.f;
    float pre = xt;
#pragma unroll
    for (int i = 0; i < HH; ++i) {
      pre += rdlane(hprev, i) * wsc[i];
      pre += rdlane(hc, i) * wcc[i];
    }
    float hn = tanhf(pre);
    ring[t & 255][j] = hn;
    hprev = hn;
    h_all[hbase + (size_t)tt * 64] = hn;
  }
}

// ---------------------------------------------------------------------------
// Kernel 3: per (b, period): HtH = Hm^T Hm (+reg), HtX = Hm^T X via
// V_WMMA_F32_16X16X4_F32, then fused 128x128 solve in LDS, W_out -> d_out.
// 8 waves; wave w owns C row-block w (8 HtH tiles + 1 HtX tile).
// Hm chunk stored ROW-PAIR INTERLEAVED: (r,c) -> pair (r>>1), addr
// pair*288 + 2c + (r&1). A lane's WMMA operand half (K,K+1 at fixed col) is
// one aligned 8B word -> single ds_load_b64 per operand, immediate offsets.
// ---------------------------------------------------------------------------
constexpr int PSTR = 288;  // floats per interleaved row-pair (288 % 64 == 32)
constexpr int XSTR = 32;   // floats per interleaved X row-pair
struct GemmBufs { float Hm2[16 * PSTR]; float Xc2[16 * XSTR]; };
struct SolveBufs { float A[128][129]; float X[128][8]; float pv[128]; float pvX[8]; };
union SmemU { GemmBufs g; SolveBufs s; };

__global__ __launch_bounds__(256) void gemm_solve_kernel(const float* __restrict__ h_all,
                                                         const float* __restrict__ x,
                                                         const float* __restrict__ regular,
                                                         float* __restrict__ out) {
  __shared__ SmemU sm;
  int bi = blockIdx.x;       // b*2 + i
  int b  = bi >> 1;
  int i  = bi & 1;
  int p  = (i == 0) ? 24 : 168;
  int T  = LL - p;
  int tid  = threadIdx.x;
  int wave = tid >> 5;
  int lane = tid & 31;
  int lo = lane & 15, hi = lane >> 4;

  v8f zero8 = {};
  v8f acc[8];
#pragma unroll
  for (int jt = 0; jt < 8; ++jt) acc[jt] = zero8;
  v8f accX = zero8;

  float* HmF = sm.g.Hm2;
  float* XcF = sm.g.Xc2;
  // per-lane bases: lane covers pair (2*ks + hi), column (block*16 + lo)
  const int hbT = hi * PSTR + lo * 2;              // B-tile base (add jt*32, ks*576)
  const int hbA = hbT + wave * 32;                 // A-tile base
  const int xbT = hi * XSTR + lo * 2;              // X-tile base (add ks*64)

  const size_t hbase = (size_t)bi * LL * 64;
  int nch = (T + 31) / 32;
  for (int ch = 0; ch < nch; ++ch) {
    int tchunk = ch * 32;
    __syncthreads();
    // stage 32 rows of Hm (128 cols), interleaving row pairs.
    // item: rp (0..15) x c4 (0,4,..,124); 512 items over 256 threads.
#pragma unroll
    for (int q = 0; q < 2; ++q) {
      int idx = tid + q * 256;          // 0..511
      int rp  = idx >> 5;               // row pair 0..15
      int c4  = (idx & 31) * 4;         // column quad
      int trA = tchunk + 2 * rp;
      int trB = trA + 1;
      float4 va = make_float4(0.f, 0.f, 0.f, 0.f);
      float4 vb = va;
      int cc = (c4 < 64) ? c4 : (c4 - 64);
      if (trA < T) {
        size_t ta = (c4 < 64) ? (size_t)(p - 1 + trA) : (size_t)trA;
        va = *(const float4*)(h_all + hbase + ta * 64 + cc);
      }
      if (trB < T) {
        size_t tb = (c4 < 64) ? (size_t)(p - 1 + trB) : (size_t)trB;
        vb = *(const float4*)(h_all + hbase + tb * 64 + cc);
      }
      float4 w0 = make_float4(va.x, vb.x, va.y, vb.y);
      float4 w1 = make_float4(va.z, vb.z, va.w, vb.w);
      *(float4*)&HmF[rp * PSTR + c4 * 2]     = w0;
      *(float4*)&HmF[rp * PSTR + c4 * 2 + 4] = w1;
    }
    // stage X chunk (32 x 8, interleaved pairs; cols 8..15 zero)
    {
      int r = tid >> 3;                 // 0..31
      int d = tid & 7;
      int tr = tchunk + r;
      float v = (tr < T) ? x[((size_t)b * LL + p + tr) * DD + d] : 0.f;
      XcF[(r >> 1) * XSTR + d * 2 + (r & 1)] = v;
      XcF[(r >> 1) * XSTR + 16 + d * 2 + (r & 1)] = 0.f;
    }
    __syncthreads();
    // 8 K-steps of 4; every operand is one ds_load_b64 with immediate offset
#pragma unroll
    for (int ks = 0; ks < 8; ++ks) {
      const int kb = ks * 2 * PSTR;
      v2f a = *(const v2f*)&HmF[hbA + kb];
      v2f tiles[8];
#pragma unroll
      for (int jt = 0; jt < 8; ++jt)
        tiles[jt] = *(const v2f*)&HmF[hbT + kb + jt * 32];
      v2f xt = *(const v2f*)&XcF[xbT + ks * 2 * XSTR];
#pragma unroll
      for (int jt = 0; jt < 8; ++jt)
        acc[jt] = __builtin_amdgcn_wmma_f32_16x16x4_f32(false, a, false, tiles[jt],
                                                        (short)0, acc[jt], false, false);
      accX = __builtin_amdgcn_wmma_f32_16x16x4_f32(false, a, false, xt,
                                                   (short)0, accX, false, false);
    }
  }
  __syncthreads();

  // scatter C tiles -> solver LDS (+ regularizer), HtX -> X
#pragma unroll
  for (int jt = 0; jt < 8; ++jt) {
#pragma unroll
    for (int r = 0; r < 8; ++r) {
      int row = wave * 16 + r + 8 * hi;
      int col = jt * 16 + lo;
      sm.s.A[row][col] = acc[jt][r] + regular[row * 128 + col];
    }
  }
  if (lo < 8) {
#pragma unroll
    for (int r = 0; r < 8; ++r) sm.s.X[wave * 16 + r + 8 * hi][lo] = accX[r];
  }
  __syncthreads();

  // Gaussian elimination (SPD: HtH + I, no pivoting needed)
  int r = tid >> 1, half = tid & 1;
  for (int c = 0; c < 127; ++c) {
    if (tid < 128) sm.s.pv[tid] = sm.s.A[c][tid];
    else if (tid < 136) sm.s.pvX[tid - 128] = sm.s.X[c][tid - 128];
    __syncthreads();
    if (r > c) {
      float f = sm.s.A[r][c] / sm.s.pv[c];
      for (int jj = c + 1 + half; jj < 128; jj += 2)
        sm.s.A[r][jj] -= f * sm.s.pv[jj];
#pragma unroll
      for (int d = 0; d < 4; ++d)
        sm.s.X[r][half * 4 + d] -= f * sm.s.pvX[half * 4 + d];
    }
    __syncthreads();
  }
  // back substitution: U W = X (in place)
  for (int c = 127; c >= 0; --c) {
    if (tid < 8) sm.s.X[c][tid] /= sm.s.A[c][c];
    __syncthreads();
    if (r < c) {
      float a = sm.s.A[r][c];
#pragma unroll
      for (int d = 0; d < 4; ++d)
        sm.s.X[r][half * 4 + d] -= a * sm.s.X[c][half * 4 + d];
    }
    __syncthreads();
  }

  if (tid < 128) {
    float* o = out + (size_t)b * FEAT + i * 1024 + tid * 8;
#pragma unroll
    for (int d = 0; d < 8; ++d) o[d] = sm.s.X[tid][d];
  }
}

// ---------------------------------------------------------------------------
// Kernel 4: max over time of h -> out[:, 2048:2176]
// ---------------------------------------------------------------------------
__global__ __launch_bounds__(64) void maxfeat_kernel(const float* __restrict__ h_all,
                                                     float* __restrict__ out) {
  int bk = blockIdx.x;      // b*2 + k
  int j  = threadIdx.x;     // 0..63
  const float* src = h_all + (size_t)bk * LL * 64 + j;
  float m = -__builtin_inff();
  for (int t = 0; t < LL; ++t) m = fmaxf(m, src[(size_t)t * 64]);
  int b = bk >> 1, k = bk & 1;
  out[(size_t)b * FEAT + 2048 + k * 64 + j] = m;
}

// ---------------------------------------------------------------------------
extern "C" void kernel_launch(void* const* d_in, const int* in_sizes, int n_in,
                              void* d_out, int out_size, void* d_ws, size_t ws_size,
                              hipStream_t stream) {
  const float* x    = (const float*)d_in[0];
  const float* W_in = (const float*)d_in[1];
  const float* W_fs = (const float*)d_in[2];
  const float* W_fc = (const float*)d_in[3];
  const float* W_bs = (const float*)d_in[4];
  const float* W_bc = (const float*)d_in[5];
  const float* regular = (const float*)d_in[6];
  float* out = (float*)d_out;
  float* ws  = (float*)d_ws;

  float* x_tr  = ws;                                  // BB*LL*HH floats (32 MB)
  float* h_all = ws + (size_t)BB * LL * HH;           // BB*2*LL*64 floats (128 MB)

  xtr_kernel<<<dim3((BB * LL * HH) / 256), dim3(256), 0, stream>>>(x, W_in, x_tr);
  scan_kernel<<<dim3(BB * 2 * 2), dim3(32), 0, stream>>>(x_tr, W_fs, W_fc, W_bs, W_bc, h_all);
  gemm_solve_kernel<<<dim3(BB * 2), dim3(256), 0, stream>>>(h_all, x, regular, out);
  maxfeat_kernel<<<dim3(BB * 2), dim3(64), 0, stream>>>(h_all, out);
}